// HIVGNN_34162169872884
// MI455X (gfx1250) — compile-verified
//
#include <hip/hip_runtime.h>
#include <hip/hip_bf16.h>

// Problem sizes (fixed by the reference)
#define NN 50000
#define EE 800000
#define FF 128
#define HH 128
#define GG 1024
#define BN_EPS 1e-5f

typedef float v2f __attribute__((ext_vector_type(2)));
typedef float v8f __attribute__((ext_vector_type(8)));

// ---------------------------------------------------------------------------
// Degree / normalization
// ---------------------------------------------------------------------------
__global__ void deg_init_k(float* deg, int n) {
    int i = blockIdx.x * 256 + threadIdx.x;
    if (i < n) deg[i] = 1.0f;               // self-loop
}

__global__ void deg_acc_k(const int* __restrict__ ei, float* deg, int e) {
    int i = blockIdx.x * 256 + threadIdx.x;
    if (i < e) atomicAdd(&deg[ei[EE + i]], 1.0f);   // dst row of edge_index
}

__global__ void deg_fin_k(float* deg, int n) {
    int i = blockIdx.x * 256 + threadIdx.x;
    if (i < n) deg[i] = rsqrtf(deg[i]);     // deg -> d^{-1/2}
}

// ---------------------------------------------------------------------------
// Node-feature GEMM: H = X(N x 128) @ W(128 x 128), W staged in LDS.
// One block = 16 output rows x 128 cols; 8 waves, one 16x16 tile per wave.
// Uses V_WMMA_F32_16X16X4_F32 (full fp32, K-step 4).
// ---------------------------------------------------------------------------
__global__ void __launch_bounds__(256)
gemm_node_k(const float* __restrict__ X, const float* __restrict__ W,
            float* __restrict__ Hout) {
    __shared__ float sW[128 * 128];         // 64 KB of the 320 KB WGP LDS
    const int tid = threadIdx.x;
    for (int i = tid; i < 128 * 128; i += 256) sW[i] = W[i];
    __syncthreads();

    const int wid  = tid >> 5;
    const int lane = tid & 31;
    const int m    = lane & 15;             // M (for A) / N (for B) index
    const int hi   = lane >> 4;             // K-pair selector
    const int row0 = blockIdx.x * 16;       // N=50000 is a multiple of 16
    const int col0 = wid * 16;

    const float* arow = X + (row0 + m) * 128 + 2 * hi;
    const float* bcol = sW + (2 * hi) * 128 + col0 + m;

    v8f c = {0.f, 0.f, 0.f, 0.f, 0.f, 0.f, 0.f, 0.f};
#pragma unroll 4
    for (int k = 0; k < 128; k += 4) {
        v2f a = *(const v2f*)(arow + k);    // A[m][k+2hi], A[m][k+2hi+1]
        v2f b;
        b.x = bcol[k * 128];                // B[k+2hi][n]
        b.y = bcol[k * 128 + 128];          // B[k+2hi+1][n]
        c = __builtin_amdgcn_wmma_f32_16x16x4_f32(
                false, a, false, b, (short)0, c, false, false);
    }
    float* outp = Hout + (row0 + 8 * hi) * 128 + col0 + m;
#pragma unroll
    for (int r = 0; r < 8; ++r) outp[r * 128] = c[r];   // M = r + 8*hi, N = m
}

// ---------------------------------------------------------------------------
// Self-loop + bias:  Agg[i,f] = H[i,f] * dinv[i]^2 + b[f]
// ---------------------------------------------------------------------------
__global__ void self_init_k(const float* __restrict__ H,
                            const float* __restrict__ dinv,
                            const float* __restrict__ bias,
                            float* __restrict__ Agg, int total) {
    int i = blockIdx.x * 256 + threadIdx.x;
    if (i >= total) return;
    int node = i >> 7, f = i & 127;
    float di = dinv[node];
    Agg[i] = H[i] * di * di + bias[f];
}

// ---------------------------------------------------------------------------
// Edge scatter: one wave per edge, 4 features per lane, fp32 global atomics.
// ---------------------------------------------------------------------------
__global__ void __launch_bounds__(256)
edge_scatter_k(const float* __restrict__ H, const int* __restrict__ ei,
               const float* __restrict__ dinv, float* __restrict__ Agg, int e) {
    int t = blockIdx.x * 256 + threadIdx.x;     // < E*32 = 25.6M
    int edge = t >> 5;
    if (edge >= e) return;
    int lane = t & 31;
    int s = ei[edge];
    int d = ei[EE + edge];
    float w = dinv[s] * dinv[d];
    const float4 hv = *(const float4*)(H + s * 128 + lane * 4);
    float* ap = Agg + d * 128 + lane * 4;
    atomicAdd(ap + 0, hv.x * w);
    atomicAdd(ap + 1, hv.y * w);
    atomicAdd(ap + 2, hv.z * w);
    atomicAdd(ap + 3, hv.w * w);
}

// ---------------------------------------------------------------------------
// BatchNorm statistics: per-channel sum / sumsq (128 channels)
// ---------------------------------------------------------------------------
__global__ void zero_k(float* p, int n) {
    int i = blockIdx.x * 256 + threadIdx.x;
    if (i < n) p[i] = 0.0f;
}

__global__ void __launch_bounds__(128)
bn_stats_k(const float* __restrict__ X, float* sum, float* sumsq, int nrows) {
    int f  = threadIdx.x;
    int r0 = blockIdx.x * 256;
    int r1 = r0 + 256; if (r1 > nrows) r1 = nrows;
    float s = 0.f, s2 = 0.f;
    for (int r = r0; r < r1; ++r) {
        float v = X[r * 128 + f];
        s += v; s2 += v * v;
    }
    atomicAdd(&sum[f], s);
    atomicAdd(&sumsq[f], s2);
}

__global__ void bn_apply_k(const float* __restrict__ X,
                           const float* __restrict__ sum,
                           const float* __restrict__ sumsq,
                           const float* __restrict__ gamma,
                           const float* __restrict__ beta,
                           float* __restrict__ Out, int total) {
    int i = blockIdx.x * 256 + threadIdx.x;
    if (i >= total) return;
    int f = i & 127;
    const float invN = 1.0f / (float)NN;
    float mu  = sum[f] * invN;
    float var = sumsq[f] * invN - mu * mu;
    float v = gamma[f] * (X[i] - mu) * rsqrtf(var + BN_EPS) + beta[f];
    Out[i] = fmaxf(v, 0.0f);                 // ReLU
}

// ---------------------------------------------------------------------------
// Pooling: counts, sums, max (bitwise uint max valid since x >= 0 post-ReLU)
// ---------------------------------------------------------------------------
__global__ void pool_cnt_k(const int* __restrict__ batch, float* cnt, int n) {
    int i = blockIdx.x * 256 + threadIdx.x;
    if (i < n) atomicAdd(&cnt[batch[i]], 1.0f);
}

__global__ void pool_scatter_k(const float* __restrict__ X,
                               const int* __restrict__ batch,
                               float* __restrict__ gsum,
                               float* __restrict__ gmax, int total) {
    int i = blockIdx.x * 256 + threadIdx.x;
    if (i >= total) return;
    int node = i >> 7, f = i & 127;
    int g = batch[node];
    float v = X[i];
    atomicAdd(&gsum[g * 128 + f], v);
    atomicMax((unsigned int*)&gmax[g * 128 + f], __float_as_uint(v));
}

__global__ void pool_cat_k(const float* __restrict__ cnt,
                           const float* __restrict__ gsum,
                           const float* __restrict__ gmax,
                           float* __restrict__ gcat, int total /* G*128 */) {
    int i = blockIdx.x * 256 + threadIdx.x;
    if (i >= total) return;
    int g = i >> 7, f = i & 127;
    float c = fmaxf(cnt[g], 1.0f);
    gcat[g * 256 + f]       = gsum[i] / c;
    gcat[g * 256 + 128 + f] = gmax[i];
}

// ---------------------------------------------------------------------------
// Generic MLP GEMM: Out(M x Ncols) = A(M x K) @ W(K x Ncols) + bias, opt ReLU
// 8 waves per block, each owns one 16x16 tile; col tiles via blockIdx.y*8+wid
// ---------------------------------------------------------------------------
__global__ void __launch_bounds__(256)
gemm_mlp_k(const float* __restrict__ A, const float* __restrict__ W,
           const float* __restrict__ bias, float* __restrict__ Out,
           int K, int Ncols, int relu) {
    const int tid  = threadIdx.x;
    const int wid  = tid >> 5;
    const int lane = tid & 31;
    const int m    = lane & 15;
    const int hi   = lane >> 4;
    const int ct   = blockIdx.y * 8 + wid;
    if (ct * 16 >= Ncols) return;            // wave-uniform exit (EXEC stays full)
    const int row0 = blockIdx.x * 16;
    const int col0 = ct * 16;

    const float* arow = A + (row0 + m) * K + 2 * hi;
    const float* bcol = W + (2 * hi) * Ncols + col0 + m;

    v8f c = {0.f, 0.f, 0.f, 0.f, 0.f, 0.f, 0.f, 0.f};
    for (int k = 0; k < K; k += 4) {
        v2f a = *(const v2f*)(arow + k);
        v2f b;
        b.x = bcol[k * Ncols];
        b.y = bcol[k * Ncols + Ncols];
        c = __builtin_amdgcn_wmma_f32_16x16x4_f32(
                false, a, false, b, (short)0, c, false, false);
    }
    float bb = bias[col0 + m];
    float* outp = Out + (row0 + 8 * hi) * Ncols + col0 + m;
#pragma unroll
    for (int r = 0; r < 8; ++r) {
        float v = c[r] + bb;
        if (relu) v = fmaxf(v, 0.0f);
        outp[r * Ncols] = v;
    }
}

// Final 128 -> 1 projection: one wave per graph, shuffle reduction
__global__ void __launch_bounds__(256)
final_dot_k(const float* __restrict__ H2, const float* __restrict__ W3,
            const float* __restrict__ b3, float* __restrict__ out, int g_total) {
    int wid = threadIdx.x >> 5, lane = threadIdx.x & 31;
    int g = blockIdx.x * 8 + wid;
    if (g >= g_total) return;
    float4 h = *(const float4*)(H2 + g * 128 + lane * 4);
    float4 w = *(const float4*)(W3 + lane * 4);
    float s = h.x * w.x + h.y * w.y + h.z * w.z + h.w * w.w;
#pragma unroll
    for (int off = 16; off > 0; off >>= 1) s += __shfl_xor(s, off, 32);
    if (lane == 0) out[g] = s + b3[0];
}

// ---------------------------------------------------------------------------
extern "C" void kernel_launch(void* const* d_in, const int* in_sizes, int n_in,
                              void* d_out, int out_size, void* d_ws, size_t ws_size,
                              hipStream_t stream) {
    const float* x     = (const float*)d_in[0];
    const int*   ei    = (const int*)  d_in[1];   // [2, E]
    const int*   batch = (const int*)  d_in[2];
    const float* W0    = (const float*)d_in[3];   // [128,128]
    const float* Wc    = (const float*)d_in[4];   // [2,128,128]
    const float* cb    = (const float*)d_in[5];   // [3,128]
    const float* gmm   = (const float*)d_in[6];   // bn_gamma [3,128]
    const float* bta   = (const float*)d_in[7];   // bn_beta  [3,128]
    const float* mW1   = (const float*)d_in[8];   // [256,256]
    const float* mb1   = (const float*)d_in[9];
    const float* mW2   = (const float*)d_in[10];  // [256,128]
    const float* mb2   = (const float*)d_in[11];
    const float* mW3   = (const float*)d_in[12];  // [128,1]
    const float* mb3   = (const float*)d_in[13];
    float* out = (float*)d_out;

    // workspace layout (floats)
    float* ws    = (float*)d_ws;
    const int NHf = NN * HH;                      // 6.4M floats
    float* B0    = ws;                            // node buffer 0
    float* B1    = B0 + NHf;                      // node buffer 1
    float* dinv  = B1 + NHf;                      // [N]
    float* bsum  = dinv + NN;                     // [128]
    float* bsq   = bsum + 128;                    // [128]  (contiguous with bsum)
    float* cnt   = bsq + 128;                     // [G]
    float* gsum  = cnt + GG;                      // [G,128]
    float* gmax  = gsum + GG * 128;               // [G,128] (contiguous w/ cnt,gsum)
    float* gcat  = gmax + GG * 128;               // [G,256]
    float* h1    = gcat + GG * 256;               // [G,256]
    float* h2    = h1 + GG * 256;                 // [G,128]
    (void)ws_size; (void)in_sizes; (void)n_in; (void)out_size;

    // ---- GCN normalization coefficients ----
    deg_init_k<<<(NN + 255) / 256, 256, 0, stream>>>(dinv, NN);
    deg_acc_k <<<(EE + 255) / 256, 256, 0, stream>>>(ei, dinv, EE);
    deg_fin_k <<<(NN + 255) / 256, 256, 0, stream>>>(dinv, NN);

    // ---- three GCN layers (ping-pong B0/B1) ----
    const float* xc = x;
    for (int layer = 0; layer < 3; ++layer) {
        const float* W  = (layer == 0) ? W0 : (Wc + (layer - 1) * 128 * 128);
        float* hb = (layer & 1) ? B1 : B0;        // GEMM output
        float* ab = (layer & 1) ? B0 : B1;        // aggregation buffer

        gemm_node_k<<<NN / 16, 256, 0, stream>>>(xc, W, hb);
        self_init_k<<<(NHf + 255) / 256, 256, 0, stream>>>(
            hb, dinv, cb + layer * 128, ab, NHf);
        edge_scatter_k<<<(EE * 32) / 256, 256, 0, stream>>>(hb, ei, dinv, ab, EE);
        zero_k<<<1, 256, 0, stream>>>(bsum, 256);  // bsum + bsq
        bn_stats_k<<<(NN + 255) / 256, 128, 0, stream>>>(ab, bsum, bsq, NN);
        bn_apply_k<<<(NHf + 255) / 256, 256, 0, stream>>>(
            ab, bsum, bsq, gmm + layer * 128, bta + layer * 128, hb, NHf);
        xc = hb;                                   // layer output
    }

    // ---- global mean/max pooling ----
    int pz = GG + 2 * GG * 128;                    // cnt + gsum + gmax contiguous
    zero_k<<<(pz + 255) / 256, 256, 0, stream>>>(cnt, pz);
    pool_cnt_k<<<(NN + 255) / 256, 256, 0, stream>>>(batch, cnt, NN);
    pool_scatter_k<<<(NHf + 255) / 256, 256, 0, stream>>>(xc, batch, gsum, gmax, NHf);
    pool_cat_k<<<(GG * 128 + 255) / 256, 256, 0, stream>>>(cnt, gsum, gmax, gcat, GG * 128);

    // ---- MLP head ----
    gemm_mlp_k<<<dim3(GG / 16, 2), 256, 0, stream>>>(gcat, mW1, mb1, h1, 256, 256, 1);
    gemm_mlp_k<<<dim3(GG / 16, 1), 256, 0, stream>>>(h1, mW2, mb2, h2, 256, 128, 1);
    final_dot_k<<<GG / 8, 256, 0, stream>>>(h2, mW3, mb3, out, GG);
}